// RollingStaticGCN_17386027614713
// MI455X (gfx1250) — compile-verified
//
#include <hip/hip_runtime.h>
#include <math.h>

typedef __attribute__((ext_vector_type(2))) float v2f;
typedef __attribute__((ext_vector_type(8))) float v8f;

#define F_IN  128
#define F_HID 128
#define F_OUT 2

// ---------------------------------------------------------------------------
// Degree / normalization kernels
// ---------------------------------------------------------------------------
__global__ void k_deg_init(unsigned* deg, int n) {
    int i = blockIdx.x * blockDim.x + threadIdx.x;
    if (i < n) deg[i] = 1u;  // self-loop
}

__global__ void k_deg_count(const int* __restrict__ dst, unsigned* deg, int e) {
    int i = blockIdx.x * blockDim.x + threadIdx.x;
    if (i < e) atomicAdd(&deg[dst[i]], 1u);
}

__global__ void k_deg_to_dinv(float* dinv, int n) {
    int i = blockIdx.x * blockDim.x + threadIdx.x;
    if (i < n) {
        unsigned d = reinterpret_cast<unsigned*>(dinv)[i];
        dinv[i] = rsqrtf((float)d);  // d >= 1 always (self-loop)
    }
}

// ---------------------------------------------------------------------------
// GEMM1: H1 = X @ W1   (50000x128 @ 128x128), fp32 WMMA 16x16x4.
// One wave computes a 16(M) x 128(N) strip: 8 accumulators, K stepped by 4.
// A layout (16x4 f32): lane m = lane&15; lanes 0-15 hold K=0,1 in vgpr0/1,
// lanes 16-31 hold K=2,3. B layout symmetric (n = lane&15).
// D layout: D[i + 8*(lane>>4)][lane&15] in accumulator element i.
// ---------------------------------------------------------------------------
__global__ __launch_bounds__(32) void k_gemm1_wmma(const float* __restrict__ X,
                                                   const float* __restrict__ W,
                                                   float* __restrict__ H,
                                                   int n_nodes) {
    const int mtile = blockIdx.x;
    const int lane  = threadIdx.x;
    const int half  = lane >> 4;
    const int lm    = lane & 15;
    int row = mtile * 16 + lm;
    if (row >= n_nodes) row = n_nodes - 1;  // defensive clamp (N%16==0 here)

    v8f c[8];
#pragma unroll
    for (int t = 0; t < 8; ++t) c[t] = v8f{0.f,0.f,0.f,0.f,0.f,0.f,0.f,0.f};

    const float* xrow = X + (long)row * F_IN;
#pragma unroll 4
    for (int k = 0; k < F_IN; k += 4) {
        const int k0 = k + 2 * half;
        v2f a;
        a.x = xrow[k0];
        a.y = xrow[k0 + 1];
#pragma unroll
        for (int t = 0; t < 8; ++t) {
            v2f b;
            b.x = W[k0 * F_HID + t * 16 + lm];
            b.y = W[(k0 + 1) * F_HID + t * 16 + lm];
            c[t] = __builtin_amdgcn_wmma_f32_16x16x4_f32(
                false, a, false, b, (short)0, c[t], false, false);
        }
    }
#pragma unroll
    for (int t = 0; t < 8; ++t)
#pragma unroll
        for (int i = 0; i < 8; ++i) {
            int m = mtile * 16 + i + 8 * half;
            if (m < n_nodes)
                H[(long)m * F_HID + t * 16 + lm] = c[t][i];
        }
}

// ---------------------------------------------------------------------------
// Layer-1 aggregation: Hagg = b1 + sum_e w_e * H1[src_e]  (+ self-loop)
// ---------------------------------------------------------------------------
__global__ void k_agg1_self(const float* __restrict__ b1,
                            const float* __restrict__ dinv,
                            const float* __restrict__ H1,
                            float* __restrict__ Hagg, int n) {
    int gid  = blockIdx.x * blockDim.x + threadIdx.x;
    int node = gid >> 5;
    int lane = gid & 31;
    if (node >= n) return;
    float w = dinv[node] * dinv[node];
    float4 v  = reinterpret_cast<const float4*>(H1 + (long)node * F_HID)[lane];
    float4 bb = reinterpret_cast<const float4*>(b1)[lane];
    float4 o;
    o.x = bb.x + w * v.x; o.y = bb.y + w * v.y;
    o.z = bb.z + w * v.z; o.w = bb.w + w * v.w;
    reinterpret_cast<float4*>(Hagg + (long)node * F_HID)[lane] = o;
}

__global__ void k_agg1_edges(const int* __restrict__ src,
                             const int* __restrict__ dst,
                             const float* __restrict__ dinv,
                             const float* __restrict__ H1,
                             float* __restrict__ Hagg, int e) {
    int gid  = blockIdx.x * blockDim.x + threadIdx.x;
    int edge = gid >> 5;
    int lane = gid & 31;
    if (edge >= e) return;
    int s = src[edge], d = dst[edge];
    float w = dinv[s] * dinv[d];
    float4 v = reinterpret_cast<const float4*>(H1 + (long)s * F_HID)[lane];
    float* o = Hagg + (long)d * F_HID + lane * 4;
    atomicAdd(o + 0, w * v.x);
    atomicAdd(o + 1, w * v.y);
    atomicAdd(o + 2, w * v.z);
    atomicAdd(o + 3, w * v.w);
}

// ---------------------------------------------------------------------------
// GEMM2: T2 = relu(Hagg) @ W2  (128 -> 2). relu fused into A loads.
// B is 128x2 padded to 128x16 with zero columns; only n<2 stored.
// ---------------------------------------------------------------------------
__global__ __launch_bounds__(32) void k_gemm2_wmma(const float* __restrict__ Hagg,
                                                   const float* __restrict__ W2,
                                                   float* __restrict__ T2,
                                                   int n_nodes) {
    const int mtile = blockIdx.x;
    const int lane  = threadIdx.x;
    const int half  = lane >> 4;
    const int lm    = lane & 15;
    int row = mtile * 16 + lm;
    if (row >= n_nodes) row = n_nodes - 1;

    v8f c = v8f{0.f,0.f,0.f,0.f,0.f,0.f,0.f,0.f};
    const float* hrow = Hagg + (long)row * F_HID;
#pragma unroll 4
    for (int k = 0; k < F_HID; k += 4) {
        const int k0 = k + 2 * half;
        v2f a;
        a.x = fmaxf(hrow[k0], 0.f);
        a.y = fmaxf(hrow[k0 + 1], 0.f);
        v2f b;
        b.x = (lm < F_OUT) ? W2[k0 * F_OUT + lm] : 0.f;
        b.y = (lm < F_OUT) ? W2[(k0 + 1) * F_OUT + lm] : 0.f;
        c = __builtin_amdgcn_wmma_f32_16x16x4_f32(
            false, a, false, b, (short)0, c, false, false);
    }
    if (lm < F_OUT) {
#pragma unroll
        for (int i = 0; i < 8; ++i) {
            int m = mtile * 16 + i + 8 * half;
            if (m < n_nodes)
                T2[(long)m * F_OUT + lm] = c[i];
        }
    }
}

// ---------------------------------------------------------------------------
// Layer-2 aggregation into d_out
// ---------------------------------------------------------------------------
__global__ void k_out_self(const float* __restrict__ b2,
                           const float* __restrict__ dinv,
                           const float* __restrict__ T2,
                           float* __restrict__ out, int n) {
    int i = blockIdx.x * blockDim.x + threadIdx.x;
    if (i >= n) return;
    float w = dinv[i] * dinv[i];
    out[i * 2 + 0] = b2[0] + w * T2[i * 2 + 0];
    out[i * 2 + 1] = b2[1] + w * T2[i * 2 + 1];
}

__global__ void k_out_edges(const int* __restrict__ src,
                            const int* __restrict__ dst,
                            const float* __restrict__ dinv,
                            const float* __restrict__ T2,
                            float* __restrict__ out, int e) {
    int i = blockIdx.x * blockDim.x + threadIdx.x;
    if (i >= e) return;
    int s = src[i], d = dst[i];
    float w = dinv[s] * dinv[d];
    atomicAdd(&out[d * 2 + 0], w * T2[s * 2 + 0]);
    atomicAdd(&out[d * 2 + 1], w * T2[s * 2 + 1]);
}

// ---------------------------------------------------------------------------
extern "C" void kernel_launch(void* const* d_in, const int* in_sizes, int n_in,
                              void* d_out, int out_size, void* d_ws, size_t ws_size,
                              hipStream_t stream) {
    const float* x    = (const float*)d_in[0];
    const int*   ei   = (const int*)d_in[1];
    const float* W1   = (const float*)d_in[2];
    const float* b1   = (const float*)d_in[3];
    const float* W2   = (const float*)d_in[4];
    const float* b2   = (const float*)d_in[5];
    float*       out  = (float*)d_out;

    const int n = in_sizes[0] / F_IN;   // 50000
    const int e = in_sizes[1] / 2;      // 600000
    const int* src = ei;
    const int* dst = ei + e;

    // workspace layout (512-byte aligned regions)
    char* ws = (char*)d_ws;
    size_t off = 0;
    float* dinv = (float*)(ws + off);            off += ((size_t)n * 4 + 511) & ~511ull;
    float* H1   = (float*)(ws + off);            off += ((size_t)n * F_HID * 4 + 511) & ~511ull;
    float* Hagg = (float*)(ws + off);            off += ((size_t)n * F_HID * 4 + 511) & ~511ull;
    float* T2   = (float*)(ws + off);            off += ((size_t)n * F_OUT * 4 + 511) & ~511ull;

    const int B = 256;
    const int mtiles = (n + 15) / 16;   // 3125

    // 1) degrees -> dinv
    k_deg_init   <<<(n + B - 1) / B, B, 0, stream>>>((unsigned*)dinv, n);
    k_deg_count  <<<(e + B - 1) / B, B, 0, stream>>>(dst, (unsigned*)dinv, e);
    k_deg_to_dinv<<<(n + B - 1) / B, B, 0, stream>>>(dinv, n);

    // 2) H1 = X @ W1 (fp32 WMMA)
    k_gemm1_wmma<<<mtiles, 32, 0, stream>>>(x, W1, H1, n);

    // 3) Hagg = b1 + self-loop + edge scatter
    k_agg1_self <<<((long)n * 32 + B - 1) / B, B, 0, stream>>>(b1, dinv, H1, Hagg, n);
    k_agg1_edges<<<((long)e * 32 + B - 1) / B, B, 0, stream>>>(src, dst, dinv, H1, Hagg, e);

    // 4) T2 = relu(Hagg) @ W2 (fp32 WMMA, relu fused)
    k_gemm2_wmma<<<mtiles, 32, 0, stream>>>(Hagg, W2, T2, n);

    // 5) out = b2 + self-loop + edge scatter
    k_out_self  <<<(n + B - 1) / B, B, 0, stream>>>(b2, dinv, T2, out, n);
    k_out_edges <<<(e + B - 1) / B, B, 0, stream>>>(src, dst, dinv, T2, out, e);
}